// GAT_MaxMargin_1_68281390072285
// MI455X (gfx1250) — compile-verified
//
#include <hip/hip_runtime.h>
#include <math.h>

typedef __attribute__((ext_vector_type(2))) float v2f;
typedef __attribute__((ext_vector_type(8))) float v8f;

#define B_    8
#define N_    512
#define IND   768
#define MEMD  300
#define HID   64
#define MROWS (B_ * N_)   /* 4096 */
#define SDIM  128         /* s_i (64, +a1_b) | s_j (64) */
#define NEGINF (-1e30f)

// ---------------------------------------------------------------------------
// float atomic max via signed/unsigned integer ordering trick
// ---------------------------------------------------------------------------
__device__ __forceinline__ void atomicMaxF(float* addr, float val) {
    if (val >= 0.0f) atomicMax((int*)addr, __float_as_int(val));
    else             atomicMin((unsigned int*)addr, __float_as_uint(val));
}

__global__ void k_init(float* mx, float* sm) {
    int t = threadIdx.x;
    if (t < B_) { mx[t] = -INFINITY; sm[t] = 0.0f; }
}

// ---------------------------------------------------------------------------
// Kernel 1: h = feature @ W_w^T + W_b     [4096 x 768] x [768 x 300]
// One wave (32 threads) per 16x16 output tile, V_WMMA_F32_16X16X4_F32.
// Out-of-range B columns (n >= 300) are address-clamped but NOT zeroed:
// WMMA column n of B only feeds column n of D, and those D columns are
// never stored -> inner loop is pure load+wmma.
// ---------------------------------------------------------------------------
__global__ __launch_bounds__(32) void k_h_gemm(const float* __restrict__ feat,
                                               const float* __restrict__ Ww,
                                               const float* __restrict__ Wb,
                                               float* __restrict__ h) {
    const int lane  = threadIdx.x;
    const int l15   = lane & 15;
    const int khalf = lane >> 4;            // 0 or 1
    const int m0    = blockIdx.y * 16;
    const int n0    = blockIdx.x * 16;

    const int n  = n0 + l15;
    const int nc = (n < MEMD) ? n : 0;      // clamp address only

    const float* arow = feat + (size_t)(m0 + l15) * IND;  // A row m = m0 + l15
    const float* brow = Ww   + (size_t)nc * IND;          // B[k,n] = Ww[n,k]

    v8f c = {0.f,0.f,0.f,0.f,0.f,0.f,0.f,0.f};
    for (int k0 = 0; k0 < IND; k0 += 4) {
        const int kb = k0 + khalf * 2;
        v2f a = *(const v2f*)(arow + kb);
        v2f b = *(const v2f*)(brow + kb);
        c = __builtin_amdgcn_wmma_f32_16x16x4_f32(false, a, false, b,
                                                  (short)0, c, false, false);
    }
    // D layout: col = l15, row = v + 8*khalf
    const int ccol = n0 + l15;
    if (ccol < MEMD) {
        const float bias  = Wb[ccol];
        const int   rbase = m0 + 8 * khalf;
#pragma unroll
        for (int v = 0; v < 8; ++v)
            h[(size_t)(rbase + v) * MEMD + ccol] = c[v] + bias;
    }
}

// ---------------------------------------------------------------------------
// Kernel 2: s = h @ [a1_wA | a1_wB]^T     [4096 x 300] x [300 x 128]
// s[:, 0:64]  = h @ a1_w[:, :300]^T + a1_b   (s_i with bias folded in)
// s[:, 64:128]= h @ a1_w[:, 300:]^T          (s_j)
// ---------------------------------------------------------------------------
__global__ __launch_bounds__(32) void k_s_gemm(const float* __restrict__ h,
                                               const float* __restrict__ a1w,
                                               const float* __restrict__ a1b,
                                               float* __restrict__ s) {
    const int lane  = threadIdx.x;
    const int l15   = lane & 15;
    const int khalf = lane >> 4;
    const int m0    = blockIdx.y * 16;
    const int n0    = blockIdx.x * 16;      // 0..112, exact (128 cols)

    const int n    = n0 + l15;              // 0..127
    const int trow = (n < HID) ? n : (n - HID);
    const int koff = (n < HID) ? 0 : MEMD;
    const float* arow = h   + (size_t)(m0 + l15) * MEMD;
    const float* brow = a1w + (size_t)trow * (2 * MEMD) + koff;

    v8f c = {0.f,0.f,0.f,0.f,0.f,0.f,0.f,0.f};
    for (int k0 = 0; k0 < MEMD; k0 += 4) {  // 300/4 = 75 iters
        const int kb = k0 + khalf * 2;
        v2f a = *(const v2f*)(arow + kb);
        v2f b = *(const v2f*)(brow + kb);
        c = __builtin_amdgcn_wmma_f32_16x16x4_f32(false, a, false, b,
                                                  (short)0, c, false, false);
    }
    const int   ccol  = n0 + l15;
    const float bias  = (ccol < HID) ? a1b[ccol] : 0.0f;
    const int   rbase = m0 + 8 * khalf;
#pragma unroll
    for (int v = 0; v < 8; ++v)
        s[(size_t)(rbase + v) * SDIM + ccol] = c[v] + bias;
}

// ---------------------------------------------------------------------------
// Kernel 3: logits[b,i,j] = mask-combine(leaky(relu(s_i+s_j) . a2w + a2b))
// 256 threads -> one 16x16 (i,j) tile; s rows staged in padded LDS.
// Also computes per-batch running max via atomicMaxF.
// ---------------------------------------------------------------------------
__global__ __launch_bounds__(256) void k_logits(const float* __restrict__ s,
                                                const float* __restrict__ a2w,
                                                const float* __restrict__ a2b,
                                                const float* __restrict__ adj,
                                                float* __restrict__ P,
                                                float* __restrict__ mx) {
    __shared__ float si[16][HID + 4];   // +4 pad: conflict-free, keeps 16B align
    __shared__ float sj[16][HID + 4];
    __shared__ float w[HID];
    __shared__ float red[256];

    const int t  = threadIdx.x;
    const int b  = blockIdx.z;
    const int i0 = blockIdx.y * 16;
    const int j0 = blockIdx.x * 16;

    // cooperative fill: 256 threads x one float4 each per matrix
    const int r = t >> 4, q = t & 15;
    *(float4*)(&si[r][q * 4]) =
        *(const float4*)(s + ((size_t)(b * N_ + i0 + r)) * SDIM + q * 4);
    *(float4*)(&sj[r][q * 4]) =
        *(const float4*)(s + ((size_t)(b * N_ + j0 + r)) * SDIM + HID + q * 4);
    if (t < HID) w[t] = a2w[t];
    __syncthreads();

    const int ti = t >> 4;   // local i
    const int tj = t & 15;   // local j
    float acc = 0.0f;
#pragma unroll 8
    for (int k = 0; k < HID; ++k) {
        float v = si[ti][k] + sj[tj][k];
        v = v > 0.0f ? v : 0.0f;                 // ReLU (inside contraction)
        acc = fmaf(v, w[k], acc);
    }
    float e = acc + a2b[0];
    e = (e >= 0.0f) ? e : 0.01f * e;             // LeakyReLU(0.01)

    const int    i   = i0 + ti, j = j0 + tj;
    const size_t idx = ((size_t)b * N_ + i) * N_ + j;
    const float  m   = adj[idx];
    const float  logit = e * m + (1.0f - m) * NEGINF;
    P[idx] = logit;

    // block max -> per-batch atomic max
    red[t] = logit;
    __syncthreads();
    for (int off = 128; off > 0; off >>= 1) {
        if (t < off) red[t] = fmaxf(red[t], red[t + off]);
        __syncthreads();
    }
    if (t == 0) atomicMaxF(&mx[b], red[0]);
}

// ---------------------------------------------------------------------------
// Kernel 4: P = exp(P - max[b]) in place; per-batch sum via atomicAdd.
// ---------------------------------------------------------------------------
__global__ __launch_bounds__(256) void k_exp(float* __restrict__ P,
                                             const float* __restrict__ mx,
                                             float* __restrict__ sm) {
    __shared__ float red[256];
    const int    b    = blockIdx.y;
    const int    t    = threadIdx.x;
    const size_t base = (size_t)b * N_ * N_ + (size_t)blockIdx.x * 1024 + t * 4;

    float4 v = *(const float4*)(P + base);
    const float m = mx[b];
    v.x = __expf(v.x - m); v.y = __expf(v.y - m);
    v.z = __expf(v.z - m); v.w = __expf(v.w - m);
    *(float4*)(P + base) = v;

    red[t] = v.x + v.y + v.z + v.w;
    __syncthreads();
    for (int off = 128; off > 0; off >>= 1) {
        if (t < off) red[t] += red[t + off];
        __syncthreads();
    }
    if (t == 0) atomicAdd(&sm[b], red[0]);
}

// ---------------------------------------------------------------------------
// Kernel 5: out[b] = (1/sum[b]) * P[b] @ h[b]    [512x512] x [512x300]
// B columns past 300 address-clamped only (their D columns are never stored).
// ---------------------------------------------------------------------------
__global__ __launch_bounds__(32) void k_out_gemm(const float* __restrict__ P,
                                                 const float* __restrict__ h,
                                                 const float* __restrict__ sm,
                                                 float* __restrict__ out) {
    const int lane  = threadIdx.x;
    const int l15   = lane & 15;
    const int khalf = lane >> 4;
    const int b     = blockIdx.z;
    const int m0    = blockIdx.y * 16;
    const int n0    = blockIdx.x * 16;

    const int n  = n0 + l15;
    const int nc = (n < MEMD) ? n : 0;      // clamp address only

    const float* arow = P + ((size_t)(b * N_ + m0 + l15)) * N_;
    const float* hb   = h + (size_t)b * N_ * MEMD;

    v8f c = {0.f,0.f,0.f,0.f,0.f,0.f,0.f,0.f};
    for (int k0 = 0; k0 < N_; k0 += 4) {    // 128 iters
        const int kb = k0 + khalf * 2;
        v2f a = *(const v2f*)(arow + kb);
        v2f bb;
        bb.x = hb[(size_t)kb * MEMD + nc];        // B[k,n] = h[b,k,n]
        bb.y = hb[(size_t)(kb + 1) * MEMD + nc];
        c = __builtin_amdgcn_wmma_f32_16x16x4_f32(false, a, false, bb,
                                                  (short)0, c, false, false);
    }
    const float inv  = 1.0f / sm[b];
    const int   ccol = n0 + l15;
    if (ccol < MEMD) {
        const int rbase = m0 + 8 * khalf;
#pragma unroll
        for (int v = 0; v < 8; ++v)
            out[((size_t)(b * N_ + rbase + v)) * MEMD + ccol] = c[v] * inv;
    }
}

// ---------------------------------------------------------------------------
// Launch
// ---------------------------------------------------------------------------
extern "C" void kernel_launch(void* const* d_in, const int* in_sizes, int n_in,
                              void* d_out, int out_size, void* d_ws, size_t ws_size,
                              hipStream_t stream) {
    const float* adj  = (const float*)d_in[0];
    const float* feat = (const float*)d_in[1];
    const float* Ww   = (const float*)d_in[2];
    const float* Wb   = (const float*)d_in[3];
    const float* a1w  = (const float*)d_in[4];
    const float* a1b  = (const float*)d_in[5];
    const float* a2w  = (const float*)d_in[6];
    const float* a2b  = (const float*)d_in[7];
    float*       out  = (float*)d_out;

    // workspace layout (floats)
    float* h  = (float*)d_ws;                 // 4096*300   = 1,228,800
    float* s  = h + (size_t)MROWS * MEMD;     // 4096*128   =   524,288
    float* P  = s + (size_t)MROWS * SDIM;     // 8*512*512  = 2,097,152
    float* mx = P + (size_t)B_ * N_ * N_;     // 8
    float* sm = mx + B_;                      // 8

    k_init<<<1, 32, 0, stream>>>(mx, sm);

    // h = feature @ Ww^T + Wb : tiles (19 n, 256 m)
    k_h_gemm<<<dim3(19, MROWS / 16), 32, 0, stream>>>(feat, Ww, Wb, h);

    // s = h @ [a1A|a1B]^T (+a1_b into s_i) : tiles (8 n, 256 m)
    k_s_gemm<<<dim3(SDIM / 16, MROWS / 16), 32, 0, stream>>>(h, a1w, a1b, s);

    // logits + per-batch max
    k_logits<<<dim3(N_ / 16, N_ / 16, B_), 256, 0, stream>>>(s, a2w, a2b, adj, P, mx);

    // exp + per-batch sum   (262144 / 1024 = 256 blocks per batch)
    k_exp<<<dim3(256, B_), 256, 0, stream>>>(P, mx, sm);

    // out = (P @ h) / sum
    k_out_gemm<<<dim3(19, N_ / 16, B_), 32, 0, stream>>>(P, h, sm, out);
}